// GraphSAGE_47596827574948
// MI455X (gfx1250) — compile-verified
//
#include <hip/hip_runtime.h>

typedef __attribute__((ext_vector_type(2))) float v2f;
typedef __attribute__((ext_vector_type(8))) float v8f;

#define NU 200000
#define NM 100000
#define NE 2000000
#define HID 64
#define EMB 32

// ---------------- utility kernels ----------------

__global__ void zero_f32(float* __restrict__ p, long n) {
    long i = (long)blockIdx.x * blockDim.x + threadIdx.x;
    long stride = (long)gridDim.x * blockDim.x;
    for (; i < n; i += stride) p[i] = 0.0f;
}

// copy nsrc elements, zero-fill up to ndst (row-pads K=21 weights to 24 rows)
__global__ void pad_tail(const float* __restrict__ src, float* __restrict__ dst,
                         int nsrc, int ndst) {
    int t = blockIdx.x * blockDim.x + threadIdx.x;
    if (t < ndst) dst[t] = (t < nsrc) ? src[t] : 0.0f;
}

// pad feat_movie [NM,21] -> [NM,24] (zero cols 21..23)
__global__ void pad_featM(const float* __restrict__ src, float* __restrict__ dst) {
    int t = blockIdx.x * blockDim.x + threadIdx.x;
    if (t >= NM * 24) return;
    int r = t / 24;
    int j = t - r * 24;
    dst[t] = (j < 21) ? src[r * 21 + j] : 0.0f;
}

// layer-1 scatter + degree count: 32 threads per edge.
// lanes 0-19: user feat -> movie sums (stride 20)
// lanes 0-20: movie feat -> user sums (stride 24, pad cols stay 0)
// lane 24/25: degree counters
__global__ void agg1_kernel(const float* __restrict__ featU, const float* __restrict__ featM,
                            const int* __restrict__ eu, const int* __restrict__ em,
                            float* __restrict__ sumU2M, float* __restrict__ sumM2U,
                            float* __restrict__ degM, float* __restrict__ degU, int E) {
    int t = blockIdx.x * blockDim.x + threadIdx.x;
    int e = t >> 5;
    int j = t & 31;
    if (e >= E) return;
    int u = eu[e];
    int m = em[e];
    if (j < 20) atomicAdd(&sumU2M[(long)m * 20 + j], featU[(long)u * 20 + j]);
    if (j < 21) atomicAdd(&sumM2U[(long)u * 24 + j], featM[(long)m * 21 + j]);
    if (j == 24) atomicAdd(&degM[m], 1.0f);
    if (j == 25) atomicAdd(&degU[u], 1.0f);
}

// layer-2 scatter: 64 threads per edge (H=64)
__global__ void agg2_kernel(const float* __restrict__ hU, const float* __restrict__ hM,
                            const int* __restrict__ eu, const int* __restrict__ em,
                            float* __restrict__ sumU2M, float* __restrict__ sumM2U, int E) {
    long t = (long)blockIdx.x * blockDim.x + threadIdx.x;
    int e = (int)(t >> 6);
    int j = (int)(t & 63);
    if (e >= E) return;
    int u = eu[e];
    int m = em[e];
    atomicAdd(&sumU2M[(long)m * HID + j], hU[(long)u * HID + j]);
    atomicAdd(&sumM2U[(long)u * HID + j], hM[(long)m * HID + j]);
}

// ---------------- WMMA GEMM (f32, 16x16x4), compile-time K ----------------
// A layout (f32 16x4): lanes 0-15: M=lane, VGPR0=K0,VGPR1=K1;
//                      lanes 16-31: M=lane-16, VGPR0=K2,VGPR1=K3.
// B mirrors: VGPR0 = row (k0+2*half), VGPR1 = row (k0+2*half+1), N = lid.
// C/D: VGPR i -> M = i + 8*half, N = lid.
// SCALE: multiply A fragment by 'ascale' (folds mean = sum/deg into the GEMM).
template <int K, int NCOLS, bool SCALE>
__device__ __forceinline__ v8f wmma_gemm_f32(const float* __restrict__ X,
                                             const float* __restrict__ W,
                                             long row, int col, int half,
                                             float ascale, v8f acc) {
    const float* xp = X + row * K + 2 * half;       // a.x index; a.y adjacent
    const float* wp = W + 2 * half * NCOLS + col;   // b.x row; b.y one row below
#pragma unroll
    for (int k0 = 0; k0 < K; k0 += 4) {
        v2f a = *(const v2f*)(xp + k0);
        if (SCALE) { a.x *= ascale; a.y *= ascale; }
        v2f b;
        b.x = wp[(long)k0 * NCOLS];
        b.y = wp[(long)(k0 + 1) * NCOLS];
        acc = __builtin_amdgcn_wmma_f32_16x16x4_f32(false, a, false, b,
                                                    (short)0, acc, false, false);
    }
    return acc;
}

// out[M x NCOLS] = (relu?)( Xself[KS] @ Wself + (Xneigh[KN]/deg) @ Wneigh + bias )
// block = (NCOLS/16) waves; each wave owns one 16x16 tile; grid.x = M/16.
template <int KS, int KN, int NCOLS, bool RELU>
__global__ void sage_gemm(const float* __restrict__ Xself,
                          const float* __restrict__ Wself,
                          const float* __restrict__ Xneigh,
                          const float* __restrict__ Wneigh,
                          const float* __restrict__ deg,
                          const float* __restrict__ bias,
                          float* __restrict__ out) {
    int wave = threadIdx.x >> 5;
    int lane = threadIdx.x & 31;
    int half = lane >> 4;
    int lid  = lane & 15;
    long row  = (long)blockIdx.x * 16 + lid;
    int  col0 = wave * 16;

    // mean factor for this dst row; deg is a small positive count, so the
    // hardware v_rcp_f32 (~1 ulp) is fine and avoids the IEEE div expansion.
    float nscale = __builtin_amdgcn_rcpf(fmaxf(deg[row], 1.0f));

    v8f acc = {};
    acc = wmma_gemm_f32<KS, NCOLS, false>(Xself,  Wself,  row, col0 + lid, half, 1.0f, acc);
    acc = wmma_gemm_f32<KN, NCOLS, true >(Xneigh, Wneigh, row, col0 + lid, half, nscale, acc);

    float bv = bias[col0 + lid];
    long obase = ((long)blockIdx.x * 16 + half * 8) * NCOLS + col0 + lid;
#pragma unroll
    for (int i = 0; i < 8; ++i) {
        float v = acc[i] + bv;
        if (RELU) v = fmaxf(v, 0.0f);
        out[obase + (long)i * NCOLS] = v;
    }
}

// ---------------- launch ----------------

extern "C" void kernel_launch(void* const* d_in, const int* in_sizes, int n_in,
                              void* d_out, int out_size, void* d_ws, size_t ws_size,
                              hipStream_t stream) {
    const float* featU = (const float*)d_in[0];   // [NU,20]
    const float* featM = (const float*)d_in[1];   // [NM,21]
    const int*   eu    = (const int*)d_in[2];     // [E]
    const int*   em    = (const int*)d_in[3];     // [E]
    const float* W_nr1 = (const float*)d_in[4];   // [20,64]
    const float* W_sr1 = (const float*)d_in[5];   // [21,64]
    const float* b_r1  = (const float*)d_in[6];   // [64]
    const float* W_nb1 = (const float*)d_in[7];   // [21,64]
    const float* W_sb1 = (const float*)d_in[8];   // [20,64]
    const float* b_b1  = (const float*)d_in[9];   // [64]
    const float* W_nr2 = (const float*)d_in[10];  // [64,32]
    const float* W_sr2 = (const float*)d_in[11];  // [64,32]
    const float* b_r2  = (const float*)d_in[12];  // [32]
    const float* W_nb2 = (const float*)d_in[13];  // [64,32]
    const float* W_sb2 = (const float*)d_in[14];  // [64,32]
    const float* b_b2  = (const float*)d_in[15];  // [32]
    float* out = (float*)d_out;                   // [NU+NM, 32]
    float* ws  = (float*)d_ws;

    // ---- workspace layout (floats), layer-1 scratch aliased with layer-2 sums ----
    // region A (19.2M): layer1 = sumU2M[NM*20] | sumM2U[NU*24] | featM24[NM*24]
    //                   layer2 = sum2M[NM*64]  | sum2U[NU*64]
    float* regionA = ws;
    float* sumU2M  = regionA;                          // NM*20 = 2.0M
    float* sumM2U  = sumU2M + (long)NM * 20;           // NU*24 = 4.8M
    float* featM24 = sumM2U + (long)NU * 24;           // NM*24 = 2.4M
    float* sum2M   = regionA;                          // NM*64 = 6.4M
    float* sum2U   = sum2M + (long)NM * HID;           // NU*64 = 12.8M
    long regionA_sz = (long)NM * HID + (long)NU * HID; // 19.2M

    float* degM = regionA + regionA_sz;                // NM
    float* degU = degM + NM;                           // NU
    float* hM   = degU + NU;                           // NM*64
    float* hU   = hM + (long)NM * HID;                 // NU*64
    float* W_sr1p = hU + (long)NU * HID;               // 24*64 (row-padded)
    float* W_nb1p = W_sr1p + 24 * HID;                 // 24*64

    // 1. zero layer-1 accumulators + degrees
    zero_f32<<<2048, 256, 0, stream>>>(sumU2M, (long)NM * 20 + (long)NU * 24);
    zero_f32<<<1024, 256, 0, stream>>>(degM, (long)(NM + NU));

    // 2. pad inputs (featM cols 21->24, K=21 weights rows 21->24)
    pad_featM<<<(NM * 24 + 255) / 256, 256, 0, stream>>>(featM, featM24);
    pad_tail<<<(24 * HID + 255) / 256, 256, 0, stream>>>(W_sr1, W_sr1p, 21 * HID, 24 * HID);
    pad_tail<<<(24 * HID + 255) / 256, 256, 0, stream>>>(W_nb1, W_nb1p, 21 * HID, 24 * HID);

    // 3. layer-1 aggregation (feature sums + degrees in one edge pass)
    agg1_kernel<<<(NE * 32) / 256, 256, 0, stream>>>(featU, featM, eu, em,
                                                     sumU2M, sumM2U, degM, degU, NE);

    // 4. layer-1 GEMMs (N=64 -> 4 waves = 128 threads; grid = rows/16), ReLU;
    //    mean = sum/deg folded into the neighbor A-operand inside the GEMM
    sage_gemm<24, 20, HID, true><<<NM / 16, 128, 0, stream>>>(
        featM24, W_sr1p, sumU2M, W_nr1, degM, b_r1, hM);
    sage_gemm<20, 24, HID, true><<<NU / 16, 128, 0, stream>>>(
        featU, W_sb1, sumM2U, W_nb1p, degU, b_b1, hU);

    // 5. layer-2: re-zero aliased region, aggregate H=64 hidden vectors
    zero_f32<<<4096, 256, 0, stream>>>(regionA, regionA_sz);
    agg2_kernel<<<(NE * 64) / 256, 256, 0, stream>>>(hU, hM, eu, em,
                                                     sum2M, sum2U, NE);

    // 6. layer-2 GEMMs (N=32 -> 2 waves = 64 threads), straight into d_out
    //    users first, then movies (reference concat order [h_u2; h_m2])
    sage_gemm<HID, HID, EMB, false><<<NU / 16, 64, 0, stream>>>(
        hU, W_sb2, sum2U, W_nb2, degU, b_b2, out);
    sage_gemm<HID, HID, EMB, false><<<NM / 16, 64, 0, stream>>>(
        hM, W_sr2, sum2M, W_nr2, degM, b_r2, out + (long)NU * EMB);
}